// ResiduesEncode_16363825397922
// MI455X (gfx1250) — compile-verified
//
#include <hip/hip_runtime.h>

// ResiduesEncode for MI455X (gfx1250, wave32, WMMA).
// Fused: delta -> W0 (K=14, scalar FMA) -> Wh layer0 -> Wh layer1, all in LDS.
// GEMM core: V_WMMA_F32_16X16X4_F32, M=256 (16 waves x 16-row strips),
// N=96 per workgroup (6 tiles of 16), K=256 (64 k-steps of 4).
// Panel is pair-interleaved over K so each B-fragment is ONE ds_load_b64.

typedef __attribute__((ext_vector_type(2))) float v2f;
typedef __attribute__((ext_vector_type(8))) float v8f;

#define B_       128
#define R_       512
#define VDIM_    256
#define MAXA_    14
#define NCOL     96      // columns per workgroup = 32 batches * 3 (xyz)
#define NTILE    6       // NCOL / 16 column tiles per wave
#define NCHUNK   4       // (B_*3) / NCOL
#define CW       112     // padded pair-row width (floats-pairs per K-pair row)
#define NTHREADS 512     // 16 wave32s

// Pair-interleaved panel addressing: element (k, col) lives at
//   panel[ ((k>>1)*CW + col)*2 + (k&1) ]
// so (k, k+1) for even k are adjacent -> 64-bit LDS accesses.
__device__ __forceinline__ int pidx(int k, int col) {
    return (((k >> 1) * CW) + col) * 2 + (k & 1);
}

__global__ __launch_bounds__(NTHREADS)
void residues_encode_kernel(const float* __restrict__ x,
                            const float* __restrict__ W0,
                            const float* __restrict__ Wh,
                            const int*   __restrict__ type_ids,
                            const int*   __restrict__ res_idx,
                            float*       __restrict__ out,
                            int n_atoms)
{
    extern __shared__ float panel[];           // (VDIM_/2)*CW*2 floats = 114688 B
    __shared__ float sdelta[MAXA_ * NCOL];     // 5376 B: 14 x 96 delta tile

    const int wg    = blockIdx.x;
    const int r     = wg / NCHUNK;
    const int chunk = wg - r * NCHUNK;
    const int tid   = threadIdx.x;
    const int t     = type_ids[r];
    const int ridx  = res_idx[r];
    const int bbase = chunk * (NCOL / 3);      // first batch index of this chunk

    // ---------------- Stage 0: delta tile + pos_ca ----------------
    for (int idx = tid; idx < MAXA_ * NCOL; idx += NTHREADS) {
        int a   = idx / NCOL;
        int col = idx - a * NCOL;
        int b   = bbase + col / 3;
        int c   = col % 3;
        int atom = ridx + a;
        atom = atom < 0 ? 0 : (atom >= n_atoms ? n_atoms - 1 : atom);
        float rootv = x[((size_t)b * n_atoms + (ridx + 1)) * 3 + c];
        sdelta[idx] = x[((size_t)b * n_atoms + atom) * 3 + c] - rootv;
    }
    if (tid < NCOL) {
        int b = bbase + tid / 3;
        int c = tid % 3;
        out[((size_t)b * R_ + r) * 3 + c] =
            x[((size_t)b * n_atoms + (ridx + 1)) * 3 + c];
    }
    __syncthreads();

    // ---------------- Stage 1: h0 = W0[t] (256x14) @ delta (14x96) ----------------
    const float* w0 = W0 + (size_t)t * VDIM_ * MAXA_;
    for (int idx = tid; idx < VDIM_ * NCOL; idx += NTHREADS) {
        int v   = idx / NCOL;
        int col = idx - v * NCOL;
        float s = 0.0f;
        #pragma unroll
        for (int a = 0; a < MAXA_; ++a)
            s = fmaf(w0[v * MAXA_ + a], sdelta[a * NCOL + col], s);
        panel[pidx(v, col)] = s;
    }
    __syncthreads();

    // ---------------- Stages 2/3: two WMMA layers, h = Wh[t][l] @ h ----------------
    const int wave = tid >> 5;                 // wave32
    const int lane = tid & 31;
    const int m0   = wave * 16;                // this wave's 16-row output strip
    const int lm   = lane & 15;                // lane's column / A-row within tile
    const int lh   = lane >> 4;                // half-wave selector for K pairing

    float* outh = out + (size_t)B_ * R_ * 3;   // h output region

    for (int l = 0; l < 2; ++l) {
        // A source: row (m0+lm) of Wh[t][l]; lane reads k = k0 + 2*lh + {0,1}
        const float* wl = Wh + (((size_t)t * 2 + l) * VDIM_ + (m0 + lm)) * VDIM_;

        v8f acc[NTILE];
        #pragma unroll
        for (int n = 0; n < NTILE; ++n) acc[n] = v8f{};

        #pragma unroll 2
        for (int k0 = 0; k0 < VDIM_; k0 += 4) {
            const int ka = k0 + 2 * lh;        // even
            // A fragment: 2 consecutive fp32 -> one global_load_b64 (L2-resident Wh)
            const v2f afrag = *(const v2f*)(wl + ka);

            // B fragments: pair (ka, ka+1) at col n0+lm -> single aligned b64 load.
            const float* prow = &panel[pidx(ka, lm)];
            #pragma unroll
            for (int n = 0; n < NTILE; ++n) {
                const v2f bfrag = *(const v2f*)(prow + 32 * n);  // +16 cols = +32 floats
                acc[n] = __builtin_amdgcn_wmma_f32_16x16x4_f32(
                    /*neg_a=*/false, afrag, /*neg_b=*/false, bfrag,
                    /*c_mod=*/(short)0, acc[n],
                    /*reuse_a=*/false, /*reuse_b=*/false);
            }
        }
        __syncthreads();   // all waves done reading panel for this layer

        if (l == 0) {
            // Write accumulators back into panel (each wave owns its row strip).
            // C/D layout: VGPR i -> row i (lanes 0-15) or row 8+i (lanes 16-31).
            #pragma unroll
            for (int i = 0; i < 8; ++i) {
                int row = m0 + i + 8 * lh;
                #pragma unroll
                for (int n = 0; n < NTILE; ++n)
                    panel[pidx(row, n * 16 + lm)] = acc[n][i];
            }
            __syncthreads();
        } else {
            // Final layer: stream accumulators straight to global h output.
            #pragma unroll
            for (int n = 0; n < NTILE; ++n) {
                int jg = chunk * NCOL + n * 16 + lm;   // global column in [0, 384)
                int b  = jg / 3;
                int c  = jg - b * 3;
                size_t base = ((size_t)b * R_ + r) * VDIM_;
                #pragma unroll
                for (int i = 0; i < 8; ++i) {
                    int v = m0 + i + 8 * lh;
                    outh[(base + v) * 3 + c] = acc[n][i];
                }
            }
        }
    }
}

extern "C" void kernel_launch(void* const* d_in, const int* in_sizes, int n_in,
                              void* d_out, int out_size, void* d_ws, size_t ws_size,
                              hipStream_t stream)
{
    const float* x        = (const float*)d_in[0];
    const float* W0       = (const float*)d_in[1];
    const float* Wh       = (const float*)d_in[2];
    const int*   type_ids = (const int*)d_in[3];
    const int*   res_idx  = (const int*)d_in[4];
    float*       out      = (float*)d_out;

    const int n_atoms = in_sizes[0] / (B_ * 3);

    const size_t shmem = (size_t)(VDIM_ / 2) * CW * 2 * sizeof(float); // 114688 B
    // Raise dynamic-LDS cap (gfx1250 WGP: 320 KB). Attribute set is idempotent,
    // not a stream op / allocation -> safe under graph capture.
    static bool attr_done = false;
    (void)hipFuncSetAttribute((const void*)residues_encode_kernel,
                              hipFuncAttributeMaxDynamicSharedMemorySize,
                              (int)shmem);
    (void)attr_done;

    dim3 grid(R_ * NCHUNK);     // 2048 workgroups
    dim3 block(NTHREADS);       // 16 wave32s
    residues_encode_kernel<<<grid, block, shmem, stream>>>(
        x, W0, Wh, type_ids, res_idx, out, n_atoms);
}